// attention_26164940767626
// MI455X (gfx1250) — compile-verified
//
#include <hip/hip_runtime.h>
#include <math.h>

typedef float v2f __attribute__((ext_vector_type(2)));
typedef float v8f __attribute__((ext_vector_type(8)));
typedef int   v4i __attribute__((vector_size(16)));   // matches builtin's V4i param

#if defined(__AMDGCN__) && __has_builtin(__builtin_amdgcn_global_load_async_to_lds_b128)
#define USE_ASYNC_LDS 1
#else
#define USE_ASYNC_LDS 0
#endif

namespace {
constexpr int kB       = 2;
constexpr int kMaxSeq  = 2048;
constexpr int kH       = 32;
constexpr int kD       = 128;
constexpr int kStart   = 128;
constexpr int kSeq     = 1920;
constexpr int kTotal   = kStart + kSeq;        // 2048
constexpr int kWaves   = 4;
constexpr int kBlock   = kWaves * 32;          // 4 wave32
constexpr int kQTile   = 16;                   // query rows per wave
constexpr int kRowsPerBlock = kWaves * kQTile; // 64
constexpr int kKT      = 16;                   // key tokens per tile
constexpr int kNT      = kTotal / kKT;         // 128 key tiles
constexpr int kKStride = 132;                  // padded row strides (dwords), bank-conflict free
constexpr int kVStride = 136;
constexpr int kPStride = 20;
constexpr float kScale = 0.08838834764831845f; // 1/sqrt(128)
}

__device__ __forceinline__ void cp16_to_lds(float* lds, const float* g) {
#if USE_ASYNC_LDS
  // CDNA5 async copy: no VGPR round trip, tracked by ASYNCcnt.
  __builtin_amdgcn_global_load_async_to_lds_b128(
      (__attribute__((address_space(1))) v4i*)g,
      (__attribute__((address_space(3))) v4i*)lds, 0, 0);
#else
  *(float4*)lds = *(const float4*)g;
#endif
}

__device__ __forceinline__ void stage_wait() {
#if USE_ASYNC_LDS
#if __has_builtin(__builtin_amdgcn_s_wait_asynccnt)
  __builtin_amdgcn_s_wait_asynccnt(0);
#else
  asm volatile("s_wait_asynccnt 0x0" ::: "memory");
#endif
#endif
}

__global__ __launch_bounds__(kBlock)
void fa_fwd_f32wmma(const float* __restrict__ q, const float* __restrict__ kx,
                    const float* __restrict__ vx, const float* __restrict__ kc,
                    const float* __restrict__ vc, float* __restrict__ out)
{
  __shared__ float sK[2][kKT * kKStride];   // double-buffered K tile
  __shared__ float sV[2][kKT * kVStride];   // double-buffered V tile
  __shared__ float sP[kWaves][kQTile * kPStride];

  const int b    = blockIdx.z;
  const int h    = blockIdx.y;
  const int tid  = threadIdx.x;
  const int wave = tid >> 5;
  const int lane = tid & 31;
  const int ln   = lane & 15;
  const int hi   = lane >> 4;

  const int qrow0 = blockIdx.x * kRowsPerBlock + wave * kQTile;
  const v8f vzero = {0.f,0.f,0.f,0.f,0.f,0.f,0.f,0.f};

  // ---- this wave's Q tile as 32 chained 16x4 fp32 A-fragments ----
  const float* qrow = q + (((size_t)b * kSeq + qrow0 + ln) * kH + h) * (size_t)kD;
  v2f aq[32];
#pragma unroll
  for (int c = 0; c < 32; ++c)
    aq[c] = *(const v2f*)(qrow + 4 * c + 2 * hi);

  v8f   acc[8];
  float mrow[8], lrow[8];
#pragma unroll
  for (int t = 0; t < 8; ++t) acc[t] = vzero;
#pragma unroll
  for (int r = 0; r < 8; ++r) { mrow[r] = -INFINITY; lrow[r] = 0.f; }

  float* pb = &sP[wave][0];

  // cooperative staging pattern: 128 threads x 16 contiguous floats per tile
  const int srow = tid >> 3;        // token-in-tile 0..15
  const int scol = (tid & 7) * 16;  // contiguous 64B chunk

  auto stage = [&](int buf, int kt) {
    const int t = kt * kKT + srow;
    const float *ksrc, *vsrc;
    if (t < kStart) {               // cached prefix
      const size_t off = (((size_t)b * kMaxSeq + t) * kH + h) * (size_t)kD;
      ksrc = kc + off; vsrc = vc + off;
    } else {                        // freshly appended K/V
      const size_t off = (((size_t)b * kSeq + (t - kStart)) * kH + h) * (size_t)kD;
      ksrc = kx + off; vsrc = vx + off;
    }
#pragma unroll
    for (int i = 0; i < 4; ++i) {
      cp16_to_lds(&sK[buf][srow * kKStride + scol + 4 * i], ksrc + scol + 4 * i);
      cp16_to_lds(&sV[buf][srow * kVStride + scol + 4 * i], vsrc + scol + 4 * i);
    }
  };

  stage(0, 0);   // prefill buffer 0

  for (int kt = 0; kt < kNT; ++kt) {
    const int cur = kt & 1;

    stage_wait();      // own async copies (issued last iteration) landed in LDS
    __syncthreads();   // all waves' copies visible; all reads of buf cur^1 done

    if (kt + 1 < kNT) stage(cur ^ 1, kt + 1);   // overlap next-tile copy with compute

    const float* bK = sK[cur];
    const float* bV = sV[cur];

    // ---- S = Q * K^T : two independent WMMA chains for ILP ----
    v8f s0 = vzero, s1 = vzero;
#pragma unroll
    for (int c = 0; c < 32; c += 2) {
      const v2f bk0 = *(const v2f*)&bK[ln * kKStride + 4 * c + 2 * hi];
      const v2f bk1 = *(const v2f*)&bK[ln * kKStride + 4 * (c + 1) + 2 * hi];
      s0 = __builtin_amdgcn_wmma_f32_16x16x4_f32(false, aq[c],     false, bk0,
                                                 (short)0, s0, false, false);
      s1 = __builtin_amdgcn_wmma_f32_16x16x4_f32(false, aq[c + 1], false, bk1,
                                                 (short)0, s1, false, false);
    }
    const v8f s = s0 + s1;

    // ---- online softmax (rows live in 16-lane groups of the C layout) ----
    float pnew[8];
#pragma unroll
    for (int r = 0; r < 8; ++r) {
      float sv = s[r] * kScale;
      float rm = sv;
      rm = fmaxf(rm, __shfl_xor(rm, 1, 32));
      rm = fmaxf(rm, __shfl_xor(rm, 2, 32));
      rm = fmaxf(rm, __shfl_xor(rm, 4, 32));
      rm = fmaxf(rm, __shfl_xor(rm, 8, 32));
      const float mo = mrow[r];
      const float mn = fmaxf(mo, rm);
      const float al = __expf(mo - mn);   // exp(-inf)=0 on first tile
      const float p  = __expf(sv - mn);
      float rs = p;
      rs += __shfl_xor(rs, 1, 32);
      rs += __shfl_xor(rs, 2, 32);
      rs += __shfl_xor(rs, 4, 32);
      rs += __shfl_xor(rs, 8, 32);
      mrow[r] = mn;
      lrow[r] = lrow[r] * al + rs;
      pnew[r] = p;
#pragma unroll
      for (int t = 0; t < 8; ++t) acc[t][r] *= al;
    }

    // park P in per-wave LDS (C layout -> row major); wave-local ordering only
#pragma unroll
    for (int r = 0; r < 8; ++r)
      pb[(r + 8 * hi) * kPStride + ln] = pnew[r];
    asm volatile("s_wait_dscnt 0x0" ::: "memory");  // DS in-order within wave

    // ---- acc += P * V ----
    v2f ap[4];
#pragma unroll
    for (int c = 0; c < 4; ++c)
      ap[c] = *(const v2f*)&pb[ln * kPStride + 4 * c + 2 * hi];
#pragma unroll
    for (int t = 0; t < 8; ++t) {
      v8f a = acc[t];
#pragma unroll
      for (int c = 0; c < 4; ++c) {
        v2f bv;
        bv.x = bV[(4 * c + 2 * hi)     * kVStride + t * 16 + ln];
        bv.y = bV[(4 * c + 2 * hi + 1) * kVStride + t * 16 + ln];
        a = __builtin_amdgcn_wmma_f32_16x16x4_f32(false, ap[c], false, bv,
                                                  (short)0, a, false, false);
      }
      acc[t] = a;
    }
  }

  // ---- epilogue: normalize and store [B,S,H,D] fp32 ----
  float inv[8];
#pragma unroll
  for (int r = 0; r < 8; ++r) inv[r] = 1.0f / lrow[r];
#pragma unroll
  for (int r = 0; r < 8; ++r) {
    const int row = qrow0 + r + 8 * hi;
    float* orow = out + (((size_t)b * kSeq + row) * kH + h) * (size_t)kD;
#pragma unroll
    for (int t = 0; t < 8; ++t)
      orow[t * 16 + ln] = acc[t][r] * inv[r];
  }
}

extern "C" void kernel_launch(void* const* d_in, const int* in_sizes, int n_in,
                              void* d_out, int out_size, void* d_ws, size_t ws_size,
                              hipStream_t stream) {
  (void)in_sizes; (void)n_in; (void)d_ws; (void)ws_size; (void)out_size;
  const float* q  = (const float*)d_in[0];
  const float* k  = (const float*)d_in[1];
  const float* v  = (const float*)d_in[2];
  const float* kc = (const float*)d_in[3];
  const float* vc = (const float*)d_in[4];
  float* out = (float*)d_out;
  dim3 grid(kSeq / kRowsPerBlock, kH, kB);
  fa_fwd_f32wmma<<<grid, kBlock, 0, stream>>>(q, k, v, kc, vc, out);
}